// AdaECELoss_52913997087022
// MI455X (gfx1250) — compile-verified
//
#include <hip/hip_runtime.h>
#include <hip/hip_bf16.h>

#define CLS   100
#define NBKT  65536

typedef float v2f __attribute__((ext_vector_type(2)));
typedef float v8f __attribute__((ext_vector_type(8)));
typedef int   v4i __attribute__((vector_size(16)));
typedef __attribute__((address_space(1))) v4i* gv4i_p;   // global int4*
typedef __attribute__((address_space(3))) v4i* lv4i_p;   // LDS int4*

// ---------- CDNA5 async global->LDS (ASYNCcnt path) ----------
__device__ __forceinline__ void async_b128(const float* g, float* l) {
#if __has_builtin(__builtin_amdgcn_global_load_async_to_lds_b128)
  __builtin_amdgcn_global_load_async_to_lds_b128((gv4i_p)g, (lv4i_p)l, 0, 0);
#else
  unsigned lofs = (unsigned)(unsigned long long)(__attribute__((address_space(3))) char*)l;
  asm volatile("global_load_async_to_lds_b128 %0, %1, off"
               :: "v"(lofs), "v"(g) : "memory");
#endif
}

__device__ __forceinline__ void wait_async0() {
#if __has_builtin(__builtin_amdgcn_s_wait_asynccnt)
  __builtin_amdgcn_s_wait_asynccnt(0);
#else
  asm volatile("s_wait_asynccnt 0" ::: "memory");
#endif
}

// ---------- K0: zero the workspace histogram region ----------
__global__ void k_zero(unsigned* p, int n) {
  int i = blockIdx.x * blockDim.x + threadIdx.x;
  if (i < n) p[i] = 0u;
}

// ---------- K1: streaming softmax-confidence + histogram ----------
// One wave per 16-row tile. Stage tile to LDS with async b128 copies,
// accumulate sum(exp) via V_WMMA_F32_16X16X4_F32 (B = ones), track max/argmax.
__global__ void __launch_bounds__(256)
k_pass1(const float* __restrict__ logits, const int* __restrict__ labels,
        unsigned* __restrict__ cnt, unsigned* __restrict__ corr,
        float* __restrict__ sconf, int nrows) {
  __shared__ float tile[8 * 16 * 104];           // 8 waves x 16 rows x 104 dwords
  const int lane = threadIdx.x & 31;
  const int wv   = threadIdx.x >> 5;
  const int row0 = (blockIdx.x * 8 + wv) * 16;
  if (row0 >= nrows) return;                     // wave-uniform exit

  float* tl = &tile[wv * 16 * 104];
  const float* gbase = logits + (size_t)row0 * CLS;

  // stage 16x100 floats (400 float4) via async copies, LDS row stride 104 dwords
  for (int i = lane; i < 400; i += 32) {
    int r = i / 25, c4 = i % 25;
    async_b128(gbase + r * CLS + c4 * 4, tl + r * 104 + c4 * 4);
  }
  wait_async0();

  const int myrow = lane & 15;
  const int hi    = lane >> 4;                   // 0: k=4c..4c+1, 1: k=4c+2..4c+3
  const float* lr = tl + myrow * 104;

  float m = -3.4e38f; int mi = 0;
  v8f acc = {0.f,0.f,0.f,0.f,0.f,0.f,0.f,0.f};
  v2f ones = {1.0f, 1.0f};

#pragma unroll
  for (int c = 0; c < 25; ++c) {
    int k0 = 4 * c + 2 * hi;
    float2 x = *(const float2*)(lr + k0);        // ds_load_b64, 8B aligned
    if (x.x > m) { m = x.x; mi = k0; }
    if (x.y > m) { m = x.y; mi = k0 + 1; }
    v2f a; a.x = __expf(x.x); a.y = __expf(x.y); // matches ISA 16x4 f32 A layout
    acc = __builtin_amdgcn_wmma_f32_16x16x4_f32(
        false, a, false, ones, (short)0, acc, false, false);
  }

  // combine max/argmax across the two half-waves handling the same row
  float om = __shfl_xor(m, 16, 32);
  int   oi = __shfl_xor(mi, 16, 32);
  if (om > m || (om == m && oi < mi)) { m = om; mi = oi; }

  // row sum from D: lanes 0-15 hold rows 0..7 in acc[0..7]; lanes 16-31 rows 8..15
  int rsel = myrow & 7;
  float s = acc[0];
  s = (rsel == 1) ? acc[1] : s;
  s = (rsel == 2) ? acc[2] : s;
  s = (rsel == 3) ? acc[3] : s;
  s = (rsel == 4) ? acc[4] : s;
  s = (rsel == 5) ? acc[5] : s;
  s = (rsel == 6) ? acc[6] : s;
  s = (rsel == 7) ? acc[7] : s;
  float os = __shfl_xor(s, 16, 32);
  bool lowerHas = (myrow < 8);
  bool iAmLower = (lane < 16);
  float rowsum  = (lowerHas == iAmLower) ? s : os;

  if (lane < 16) {                               // one vote per row
    float conf = __expf(m) / rowsum;             // top-1 softmax confidence
    int lab = labels[row0 + myrow];
    unsigned b = (unsigned)(conf * (float)NBKT);
    if (b > NBKT - 1) b = NBKT - 1;
    atomicAdd(&cnt[b], 1u);
    if (mi == lab) atomicAdd(&corr[b], 1u);
    atomicAdd(&sconf[b], conf);
  }
}

// ---------- K2: single-block scan of the histogram + quantile boundary emit ----
__global__ void __launch_bounds__(256)
k_scan(const unsigned* __restrict__ cnt, const unsigned* __restrict__ corr,
       const float* __restrict__ sconf, float* __restrict__ Fc,
       float* __restrict__ Fa, int W) {
  __shared__ unsigned wsC[8], wsK[8];
  __shared__ double wsS[8];
  const int tid = threadIdx.x, lane = tid & 31, wv = tid >> 5;
  unsigned carryC = 0, carryK = 0; double carryS = 0.0;

  for (int chunk = 0; chunk < NBKT / 256; ++chunk) {
    int idx = chunk * 256 + tid;
    unsigned c = cnt[idx], k = corr[idx];
    double   s = (double)sconf[idx];

    unsigned iC = c, iK = k; double iS = s;       // wave inclusive scan
    for (int off = 1; off < 32; off <<= 1) {
      unsigned tC = __shfl_up(iC, off, 32);
      unsigned tK = __shfl_up(iK, off, 32);
      double   tS = __shfl_up(iS, off, 32);
      if (lane >= off) { iC += tC; iK += tK; iS += tS; }
    }
    if (lane == 31) { wsC[wv] = iC; wsK[wv] = iK; wsS[wv] = iS; }
    __syncthreads();

    unsigned pC = 0, pK = 0; double pS = 0.0;
    for (int w = 0; w < 8; ++w)
      if (w < wv) { pC += wsC[w]; pK += wsK[w]; pS += wsS[w]; }

    unsigned inclC = carryC + pC + iC;
    unsigned exclC = inclC - c;
    unsigned exclK = carryK + pK + iK - k;
    double   exclS = carryS + pS + iS - s;

    if (c > 0) {                                  // quantile boundaries in this bucket
      unsigned j0 = exclC / (unsigned)W + 1u;
      unsigned j1 = inclC / (unsigned)W;
      for (unsigned j = j0; j <= j1 && j <= 20u; ++j) {
        float frac = (float)(j * (unsigned)W - exclC) / (float)c;
        Fc[j] = (float)(exclS + (double)frac * s);
        Fa[j] = (float)exclK + frac * (float)k;
      }
    }

    unsigned totC = 0, totK = 0; double totS = 0.0;
    for (int w = 0; w < 8; ++w) { totC += wsC[w]; totK += wsK[w]; totS += wsS[w]; }
    carryC += totC; carryK += totK; carryS += totS;
    __syncthreads();
  }
}

// ---------- K3: bins -> [ece, acc_bins(20)] ----------
__global__ void k_final(const float* __restrict__ Fc, const float* __restrict__ Fa,
                        float* __restrict__ out, int W, int nrows) {
  int j = threadIdx.x;
  float d = 0.0f;
  float invW = 1.0f / (float)W;
  if (j < 20) {
    float cb = (Fc[j + 1] - Fc[j]) * invW;
    float ab = (Fa[j + 1] - Fa[j]) * invW;
    out[1 + j] = ab;
    d = fabsf(cb - ab);
  }
  for (int off = 16; off > 0; off >>= 1) d += __shfl_xor(d, off, 32);
  if (j == 0) out[0] = d * ((float)W / (float)nrows);
}

extern "C" void kernel_launch(void* const* d_in, const int* in_sizes, int n_in,
                              void* d_out, int out_size, void* d_ws, size_t ws_size,
                              hipStream_t stream) {
  (void)n_in; (void)out_size; (void)ws_size;
  const float* logits = (const float*)d_in[0];
  const int*   labels = (const int*)d_in[1];
  const int    nrows  = in_sizes[1];             // N (labels count)

  unsigned* cnt   = (unsigned*)d_ws;
  unsigned* corrp = cnt + NBKT;
  float*    sconf = (float*)(corrp + NBKT);
  float*    Fc    = (float*)((unsigned*)d_ws + 3 * NBKT);
  float*    Fa    = Fc + 32;
  float*    out   = (float*)d_out;

  int zwords = 3 * NBKT + 64;
  hipLaunchKernelGGL(k_zero, dim3((zwords + 255) / 256), dim3(256), 0, stream,
                     (unsigned*)d_ws, zwords);

  int ntiles = nrows / 16;
  int blocks = (ntiles + 7) / 8;
  hipLaunchKernelGGL(k_pass1, dim3(blocks), dim3(256), 0, stream,
                     logits, labels, cnt, corrp, sconf, nrows);

  int W = nrows / 20;
  hipLaunchKernelGGL(k_scan, dim3(1), dim3(256), 0, stream, cnt, corrp, sconf, Fc, Fa, W);
  hipLaunchKernelGGL(k_final, dim3(1), dim3(32), 0, stream, Fc, Fa, out, W, nrows);
}